// GATSCalibrator_3109556322400
// MI455X (gfx1250) — compile-verified
//
#include <hip/hip_runtime.h>
#include <math.h>

#define CC 64
#define HH 8
#define NEG_SLOPE 0.2f
#define TPW 4  // WMMA tiles (of 16 nodes) per wave

typedef float v2f __attribute__((ext_vector_type(2)));
typedef float v8f __attribute__((ext_vector_type(8)));

__device__ __forceinline__ float waveSum(float v) {
#pragma unroll
  for (int m = 16; m >= 1; m >>= 1) v += __shfl_xor(v, m, 32);
  return v;
}
__device__ __forceinline__ float waveMin(float v) {
#pragma unroll
  for (int m = 16; m >= 1; m >>= 1) v = fminf(v, __shfl_xor(v, m, 32));
  return v;
}
__device__ __forceinline__ float waveMax(float v) {
#pragma unroll
  for (int m = 16; m >= 1; m >>= 1) v = fmaxf(v, __shfl_xor(v, m, 32));
  return v;
}

__device__ __forceinline__ float atomicMaxFloat(float* addr, float v) {
  if (v >= 0.0f)
    return __int_as_float(atomicMax((int*)addr, __float_as_int(v)));
  else
    return __uint_as_float(atomicMin((unsigned int*)addr, __float_as_uint(v)));
}

// ---------------- init accumulators + zero-padded W [16 x 64] ----------------
__global__ void k_init(float* __restrict__ deg, float* __restrict__ emax,
                       float* __restrict__ esum, float* __restrict__ dconf,
                       float* __restrict__ sim,
                       const float* __restrict__ W, float* __restrict__ wpad,
                       int n) {
  int i = blockIdx.x * blockDim.x + threadIdx.x;
  if (i < n * HH) sim[i] = 0.0f;
  if (i < n) {
    deg[i]   = 0.0f;
    emax[i]  = -INFINITY;
    esum[i]  = 0.0f;
    dconf[i] = 0.0f;
  }
  if (i < 16 * CC) wpad[i] = (i < HH * CC) ? W[i] : 0.0f;
}

// ---------------- per-node: normalize, sort(64), conf, a_cluster ----------------
// one wave (32 lanes) per node; 2 elements per lane; bitonic sort via shuffles
__global__ void k_node_prep(const float* __restrict__ x,
                            const int* __restrict__ dist,
                            const float* __restrict__ train_a,
                            const float* __restrict__ dist1_a,
                            float* __restrict__ xs,
                            float* __restrict__ conf,
                            float* __restrict__ acl,
                            int n) {
  int lane = threadIdx.x & 31;
  int node = blockIdx.x * (blockDim.x >> 5) + (threadIdx.x >> 5);
  if (node >= n) return;
  const float* row = x + (size_t)node * CC;
  float v0 = row[lane];
  float v1 = row[lane + 32];

  float mn = waveMin(fminf(v0, v1));
  float mx = waveMax(fmaxf(v0, v1));

  // conf = max softmax = 1 / sum(exp(x - max))
  float s = expf(v0 - mx) + expf(v1 - mx);
  s = waveSum(s);

  float inv = 1.0f / (mx - mn + 1e-8f);
  v0 = (v0 - mn) * inv;
  v1 = (v1 - mn) * inv;

  // bitonic sort of 64 elements: index i0 = lane, i1 = lane + 32
#pragma unroll
  for (int k = 2; k <= 64; k <<= 1) {
#pragma unroll
    for (int j = k >> 1; j >= 1; j >>= 1) {
      if (j >= 32) {
        // only k==64, j==32: in-lane compare; ascending ((i & 64)==0 always)
        float lo = fminf(v0, v1), hi = fmaxf(v0, v1);
        v0 = lo;
        v1 = hi;
      } else {
        bool lower = (lane & j) == 0;
        float p0 = __shfl_xor(v0, j, 32);
        float p1 = __shfl_xor(v1, j, 32);
        bool up0 = ((lane & k) == 0);
        bool up1 = (((lane + 32) & k) == 0);
        v0 = (lower == up0) ? fminf(v0, p0) : fmaxf(v0, p0);
        v1 = (lower == up1) ? fminf(v1, p1) : fmaxf(v1, p1);
      }
    }
  }

  xs[(size_t)node * CC + lane] = v0;
  xs[(size_t)node * CC + 32 + lane] = v1;
  if (lane == 0) {
    conf[node] = 1.0f / s;
    int d = dist[node];
    acl[node] = (d == 0) ? train_a[0] : ((d == 1) ? dist1_a[0] : 1.0f);
  }
}

// ---------------- temp = x_sorted @ W^T via V_WMMA_F32_16X16X4_F32 ----------------
// B (padded 16-col W^T) hoisted into registers once per wave; TPW tiles per wave
__global__ void k_temp_wmma(const float* __restrict__ xs,
                            const float* __restrict__ wpad,  // [16][CC], rows 8..15 zero
                            const float* __restrict__ acl,
                            float* __restrict__ temp_s,
                            int ntiles, int n) {
  int lane = threadIdx.x & 31;
  int wv = blockIdx.x * (blockDim.x >> 5) + (threadIdx.x >> 5);
  int tile0 = wv * TPW;
  if (tile0 >= ntiles) return;  // wave-uniform

  int m = lane & 15;             // A: row M within tile; B: col N
  int khalf = (lane >> 4) << 1;  // 0 (lanes 0-15) or 2 (lanes 16-31)

  // Hoist all 16 B fragments (B identical for every tile): unconditional b64 loads
  const float* brow = wpad + (size_t)m * CC + khalf;
  v2f b[16];
#pragma unroll
  for (int st = 0; st < 16; ++st) {
    b[st].x = brow[st * 4];
    b[st].y = brow[st * 4 + 1];
  }

#pragma unroll
  for (int t = 0; t < TPW; ++t) {
    int tile = tile0 + t;
    if (tile >= ntiles) break;  // wave-uniform
    int node0 = tile * 16;
    const float* arow = xs + (size_t)(node0 + m) * CC + khalf;

    v8f c = {0.f, 0.f, 0.f, 0.f, 0.f, 0.f, 0.f, 0.f};
#pragma unroll
    for (int st = 0; st < 16; ++st) {
      v2f a;
      a.x = arow[st * 4];
      a.y = arow[st * 4 + 1];
      c = __builtin_amdgcn_wmma_f32_16x16x4_f32(false, a, false, b[st],
                                                (short)0, c, false, false);
    }

    // C/D layout: VGPR v -> M = v (lanes 0-15) or v+8 (lanes 16-31), N = lane&15
    int head = lane & 15;
    int mbase = (lane < 16) ? 0 : 8;
    if (head < HH) {
#pragma unroll
      for (int v = 0; v < 8; ++v) {
        int node = node0 + mbase + v;
        if (node < n) temp_s[(size_t)node * HH + head] = c[v] * acl[node];
      }
    }
  }
}

// ---------------- edge pass 1: score, seg-max, deg, dconf ----------------
__global__ void k_edge1(const float* __restrict__ x,
                        const int* __restrict__ src, const int* __restrict__ dst,
                        const float* __restrict__ acl,
                        const float* __restrict__ conf,
                        float* __restrict__ e_ws, float* __restrict__ emax,
                        float* __restrict__ deg, float* __restrict__ dconf,
                        int ne) {
  int e = blockIdx.x * blockDim.x + threadIdx.x;
  if (e >= ne) return;
  int s = src[e], d = dst[e];
  const float4* xs4 = (const float4*)(x + (size_t)s * CC);
  const float4* xd4 = (const float4*)(x + (size_t)d * CC);
  float acc = 0.0f;
#pragma unroll
  for (int i = 0; i < CC / 4; ++i) {
    float4 a = xs4[i], b = xd4[i];
    acc += a.x * b.x + a.y * b.y + a.z * b.z + a.w * b.w;
  }
  acc /= (acl[s] * acl[d]);                         // <alpha_j, alpha_i>
  float ev = (acc > 0.0f) ? acc : NEG_SLOPE * acc;  // leaky relu
  e_ws[e] = ev;
  atomicMaxFloat(&emax[d], ev);
  atomicAdd(&deg[s], 1.0f);
  atomicAdd(&dconf[d], conf[d] - conf[s]);
}

// ---------------- edge pass 2: exp + seg-sum ----------------
__global__ void k_edge2(const int* __restrict__ dst,
                        const float* __restrict__ emax,
                        float* __restrict__ e_ws, float* __restrict__ esum,
                        int ne) {
  int e = blockIdx.x * blockDim.x + threadIdx.x;
  if (e >= ne) return;
  int d = dst[e];
  float ee = expf(e_ws[e] - emax[d]);
  e_ws[e] = ee;
  atomicAdd(&esum[d], ee);
}

// ---------------- edge pass 3: attn-weighted scatter of temp_s ----------------
__global__ void k_edge3(const int* __restrict__ src, const int* __restrict__ dst,
                        const float* __restrict__ e_ws,
                        const float* __restrict__ esum,
                        const float* __restrict__ temp_s,
                        float* __restrict__ sim, int ne) {
  int e = blockIdx.x * blockDim.x + threadIdx.x;
  if (e >= ne) return;
  int s = src[e], d = dst[e];
  float attn = e_ws[e] / (esum[d] + 1e-16f);
#pragma unroll
  for (int h = 0; h < HH; ++h)
    atomicAdd(&sim[(size_t)d * HH + h], temp_s[(size_t)s * HH + h] * attn);
}

// ---------------- finalize: softplus, mean over heads, + bias ----------------
__global__ void k_final(const float* __restrict__ sim,
                        const float* __restrict__ dconf,
                        const float* __restrict__ deg,
                        const float* __restrict__ conf_coef,
                        const float* __restrict__ bias,
                        float* __restrict__ out, int n) {
  int i = blockIdx.x * blockDim.x + threadIdx.x;
  if (i >= n) return;
  float dv = deg[i];
  float dinv = (dv > 0.0f) ? 1.0f / dv : 0.0f;
  float addv = conf_coef[0] * dconf[i] * dinv;
  float s = 0.0f;
#pragma unroll
  for (int h = 0; h < HH; ++h) {
    float xx = sim[(size_t)i * HH + h] + addv;
    s += (xx > 20.0f) ? xx : log1pf(expf(xx));
  }
  out[i] = s * (1.0f / HH) + bias[0];
}

extern "C" void kernel_launch(void* const* d_in, const int* in_sizes, int n_in,
                              void* d_out, int out_size, void* d_ws, size_t ws_size,
                              hipStream_t stream) {
  const float* x        = (const float*)d_in[0];  // [N, 64]
  const float* W_temp   = (const float*)d_in[1];  // [8, 64]
  const float* conf_co  = (const float*)d_in[2];  // [1]
  const float* bias     = (const float*)d_in[3];  // [1]
  const float* train_a  = (const float*)d_in[4];  // [1]
  const float* dist1_a  = (const float*)d_in[5];  // [1]
  const int*   eidx     = (const int*)d_in[6];    // [2, E]
  const int*   dist     = (const int*)d_in[7];    // [N]

  const int n  = in_sizes[7];
  const int ne = in_sizes[6] / 2;
  const int* src = eidx;
  const int* dst = eidx + ne;

  // workspace layout (floats)
  float* ws = (float*)d_ws;
  size_t off = 0;
  float* xs     = ws + off; off += (size_t)n * CC;  // sorted normalized rows
  float* temp_s = ws + off; off += (size_t)n * HH;
  float* conf   = ws + off; off += n;
  float* acl    = ws + off; off += n;
  float* deg    = ws + off; off += n;
  float* emax   = ws + off; off += n;
  float* esum   = ws + off; off += n;
  float* dconf  = ws + off; off += n;
  float* sim    = ws + off; off += (size_t)n * HH;
  float* e_ws   = ws + off; off += ne;
  float* wpad   = ws + off; off += 16 * CC;         // zero-padded W [16][64]
  float* out    = (float*)d_out;

  const int TB = 256;
  k_init<<<(n * HH + TB - 1) / TB, TB, 0, stream>>>(deg, emax, esum, dconf, sim,
                                                    W_temp, wpad, n);

  // 8 waves per block -> 8 nodes per block
  k_node_prep<<<(n + 7) / 8, TB, 0, stream>>>(x, dist, train_a, dist1_a,
                                              xs, conf, acl, n);

  // 8 waves per block, TPW tiles per wave
  int ntiles = (n + 15) / 16;
  int nwaves = (ntiles + TPW - 1) / TPW;
  k_temp_wmma<<<(nwaves + 7) / 8, TB, 0, stream>>>(xs, wpad, acl, temp_s,
                                                   ntiles, n);

  k_edge1<<<(ne + TB - 1) / TB, TB, 0, stream>>>(x, src, dst, acl, conf,
                                                 e_ws, emax, deg, dconf, ne);
  k_edge2<<<(ne + TB - 1) / TB, TB, 0, stream>>>(dst, emax, e_ws, esum, ne);
  k_edge3<<<(ne + TB - 1) / TB, TB, 0, stream>>>(src, dst, e_ws, esum, temp_s,
                                                 sim, ne);
  k_final<<<(n + TB - 1) / TB, TB, 0, stream>>>(sim, dconf, deg, conf_co, bias,
                                                out, n);
}